// Sample_Ranking_Loss_68023692034621
// MI455X (gfx1250) — compile-verified
//
#include <hip/hip_runtime.h>
#include <hip/hip_bf16.h>
#include <stdint.h>

// ApproxNDCG ranking loss for preds/labels [2048, 64] (list_len x batch).
//
// rank_i is permutation-invariant => DCG needs no sort; IDCG positions come
// from an O(n^2) stable-descending count => no sort at all. Two n^2 passes
// over an LDS-resident column, TRANS/VALU-bound (inputs are 1MB, trivially
// cache-resident vs 23.3 TB/s HBM / 192MB L2).
//
// CDNA5 features used:
//  * global_load_async_to_lds_b32 (+ s_wait_asynccnt) staging (ASYNCcnt path).
//  * v_wmma_f32_16x16x4_f32 producing 16x16 tiles of z = alpha*(x_j - x_i)
//    directly in the C/D accumulator layout; the -alpha*x_i term is a
//    loop-invariant WMMA hoisted out of the j loop and fed back via C.
//  * v_rcp_f32 (fast reciprocal) for the sigmoid instead of IEEE division.

typedef __attribute__((ext_vector_type(2))) float v2f;
typedef __attribute__((ext_vector_type(8))) float v8f;

#define LIST_LEN 2048
#define BATCH    64
#define SPLITS   4          // row-splits per batch -> 256 blocks total
#define ROWS_PER_SPLIT (LIST_LEN / SPLITS)   // 512
#define ALPHA_F  100.0f

#if defined(__gfx1250__) && __has_builtin(__builtin_amdgcn_wmma_f32_16x16x4_f32)
#define USE_WMMA_F32_16X16X4 1
#endif

__device__ __forceinline__ unsigned lds_off(const void* p) {
    // Generic (flat) pointers into LDS carry the DS-relative byte offset in
    // their low 32 bits (flat addr = {SHARED_BASE, lds_offset}).
    return (unsigned)(uintptr_t)p;
}

__device__ __forceinline__ float fast_sigmoid(float z) {
    // 1 / (1 + exp(-z)); v_rcp_f32 instead of IEEE divide (hot path: 268M x).
    return __builtin_amdgcn_rcpf(1.0f + __expf(-z));
}

__global__ __launch_bounds__(256)
void ndcg_partial_kernel(const float* __restrict__ preds,
                         const float* __restrict__ labels,
                         float* __restrict__ partials /* [BATCH*SPLITS][2] */) {
    __shared__ float s_preds[LIST_LEN];
    __shared__ float s_labels[LIST_LEN];
    __shared__ float s_red[512];

    const int t     = threadIdx.x;           // 0..255
    const int blk   = blockIdx.x;            // 0..255
    const int b     = blk >> 2;              // batch column
    const int split = blk & (SPLITS - 1);    // row split

    // ---- stage column b of preds/labels into LDS with CDNA5 async copies ----
    #pragma unroll
    for (int k = 0; k < LIST_LEN / 256; ++k) {
        const int i = t + k * 256;
        const unsigned goff = (unsigned)((i * BATCH + b) * 4);  // strided column
        const unsigned lp = lds_off(&s_preds[i]);
        const unsigned ll = lds_off(&s_labels[i]);
        asm volatile("global_load_async_to_lds_b32 %0, %1, %2"
                     :: "v"(lp), "v"(goff), "s"(preds) : "memory");
        asm volatile("global_load_async_to_lds_b32 %0, %1, %2"
                     :: "v"(ll), "v"(goff), "s"(labels) : "memory");
    }
    asm volatile("s_wait_asynccnt 0" ::: "memory");
    __syncthreads();

    const int wave = t >> 5;                 // 0..7
    const int lane = t & 31;
    const int half = lane >> 4;              // 0 or 1
    const int ln   = lane & 15;

    // ---- pass 1: approx ranks (sigmoid pairwise sum) -> DCG partial ----
    float dcg_acc = 0.0f;
    const int itile_lo = split * (ROWS_PER_SPLIT / 16);      // 32 tiles/split
    const int itile_hi = itile_lo + (ROWS_PER_SPLIT / 16);
    for (int itile = itile_lo + wave; itile < itile_hi; itile += 8) {
        const int i0 = itile * 16;
        const float xi = s_preds[i0 + ln];   // A-matrix value, M = lane&15

        float racc[8];
        #pragma unroll
        for (int r = 0; r < 8; ++r) racc[r] = 0.0f;

#ifdef USE_WMMA_F32_16X16X4
        // Operands replicated across both A/B VGPRs (all 4 K slots) so the
        // result is independent of the K<->(vgpr,half) mapping:
        //   D1       = sum_k xi * (-alpha/4)  = -alpha*xi   (j-loop invariant)
        //   z = D2   = D1 + sum_k 0.25*(alpha*xj) = alpha*(xj - xi)
        v2f aXi;   aXi.x = xi;                 aXi.y = xi;
        v2f bNegA; bNegA.x = -0.25f * ALPHA_F; bNegA.y = -0.25f * ALPHA_F;
        v2f aQ;    aQ.x = 0.25f;               aQ.y = 0.25f;
        v8f c0 = {};
        const v8f d1 = __builtin_amdgcn_wmma_f32_16x16x4_f32(
            false, aXi, false, bNegA, 0, c0, false, false);   // hoisted
#endif
        #pragma unroll 2
        for (int jt = 0; jt < LIST_LEN / 16; ++jt) {
            const float xj = s_preds[jt * 16 + ln];  // N = lane&15
#ifdef USE_WMMA_F32_16X16X4
            v2f bXj; bXj.x = ALPHA_F * xj; bXj.y = ALPHA_F * xj;
            v8f z = __builtin_amdgcn_wmma_f32_16x16x4_f32(
                false, aQ, false, bXj, 0, d1, false, false);
            #pragma unroll
            for (int r = 0; r < 8; ++r) {
                racc[r] += fast_sigmoid(z[r]);
            }
#else
            // VALU fallback matching the C/D layout: element (r, lane) is
            // row M = r + 8*half, col N = ln.
            #pragma unroll
            for (int r = 0; r < 8; ++r) {
                const float xim = s_preds[i0 + r + 8 * half];
                racc[r] += fast_sigmoid(ALPHA_F * (xj - xim));
            }
#endif
        }

        // C/D layout: VGPR r holds row M=r (lanes 0-15) / M=r+8 (lanes 16-31),
        // with N = lane&15.  Sum over N = 16-lane reduction within each half.
        #pragma unroll
        for (int r = 0; r < 8; ++r) {
            float v = racc[r];
            v += __shfl_xor(v, 1, 32);
            v += __shfl_xor(v, 2, 32);
            v += __shfl_xor(v, 4, 32);
            v += __shfl_xor(v, 8, 32);
            if (ln == 0) {
                const int i = i0 + r + 8 * half;
                const float rank = v + 0.5f;               // includes j==i term
                const float gain = exp2f(s_labels[i]) - 1.0f;
                dcg_acc += gain / __log2f(rank + 1.0f);    // cold path: precise
            }
        }
    }

    // ---- pass 2: stable-descending positions of labels -> IDCG partial ----
    float idcg_acc = 0.0f;
    {
        const int base = split * ROWS_PER_SPLIT;   // 512 rows -> 2 per thread
        const float li0 = s_labels[base + t];
        const float li1 = s_labels[base + 256 + t];
        int cnt0 = 0, cnt1 = 0;
        const int i0 = base + t;
        const int i1 = base + 256 + t;
        #pragma unroll 4
        for (int j = 0; j < LIST_LEN; ++j) {
            const float lj = s_labels[j];
            cnt0 += (lj > li0) + ((lj == li0) & (j < i0));
            cnt1 += (lj > li1) + ((lj == li1) & (j < i1));
        }
        idcg_acc += (exp2f(li0) - 1.0f) / __log2f((float)cnt0 + 2.0f);
        idcg_acc += (exp2f(li1) - 1.0f) / __log2f((float)cnt1 + 2.0f);
    }

    // ---- block reduction of (dcg, idcg) partials ----
    s_red[t]       = dcg_acc;
    s_red[256 + t] = idcg_acc;
    __syncthreads();
    for (int s = 128; s > 0; s >>= 1) {
        if (t < s) {
            s_red[t]       += s_red[t + s];
            s_red[256 + t] += s_red[256 + t + s];
        }
        __syncthreads();
    }
    if (t == 0) {
        partials[blk * 2 + 0] = s_red[0];
        partials[blk * 2 + 1] = s_red[256];
    }
}

__global__ __launch_bounds__(64)
void ndcg_finalize_kernel(const float* __restrict__ partials,
                          float* __restrict__ out) {
    __shared__ float s[BATCH];
    const int t = threadIdx.x;              // one thread per batch
    float dcg = 0.0f, idcg = 0.0f;
    #pragma unroll
    for (int k = 0; k < SPLITS; ++k) {      // fixed combine order: deterministic
        dcg  += partials[(t * SPLITS + k) * 2 + 0];
        idcg += partials[(t * SPLITS + k) * 2 + 1];
    }
    s[t] = (idcg != 0.0f) ? (1.0f - dcg / idcg) : 0.0f;
    __syncthreads();
    for (int k = 32; k > 0; k >>= 1) {
        if (t < k) s[t] += s[t + k];
        __syncthreads();
    }
    if (t == 0) out[0] = s[0];
}

extern "C" void kernel_launch(void* const* d_in, const int* in_sizes, int n_in,
                              void* d_out, int out_size, void* d_ws, size_t ws_size,
                              hipStream_t stream) {
    const float* preds  = (const float*)d_in[0];   // [2048, 64] f32
    const float* labels = (const float*)d_in[1];   // [2048, 64] f32
    float* partials = (float*)d_ws;                // BATCH*SPLITS*2 floats

    ndcg_partial_kernel<<<BATCH * SPLITS, 256, 0, stream>>>(preds, labels, partials);
    ndcg_finalize_kernel<<<1, BATCH, 0, stream>>>(partials, (float*)d_out);
}